// DomainAlignmentModel_71614284693828
// MI455X (gfx1250) — compile-verified
//
#include <hip/hip_runtime.h>

#define NNODES 50000
#define NEDGES 800000
#define DIN    128
#define HDIM   64
#define NHSZ   (NNODES * HDIM)

typedef __attribute__((ext_vector_type(2)))  float  v2f;
typedef __attribute__((ext_vector_type(8)))  float  v8f;

// ---------------- graph normalization ----------------

__global__ void k_deg_init(float* __restrict__ deg, int n) {
    int i = blockIdx.x * blockDim.x + threadIdx.x;
    if (i < n) deg[i] = 1.0f;                 // self-loop contribution
}

__global__ void k_deg_count(const int* __restrict__ col, float* __restrict__ deg, int e) {
    int i = blockIdx.x * blockDim.x + threadIdx.x;
    if (i < e) atomicAdd(&deg[col[i]], 1.0f);
}

__global__ void k_dinv(float* __restrict__ deg, int n) {
    int i = blockIdx.x * blockDim.x + threadIdx.x;
    if (i < n) deg[i] = rsqrtf(deg[i]);       // deg >= 1 always
}

__global__ void k_edge_w(const int* __restrict__ row, const int* __restrict__ col,
                         const float* __restrict__ dinv, float* __restrict__ w, int e) {
    int i = blockIdx.x * blockDim.x + threadIdx.x;
    if (i < e) w[i] = dinv[row[i]] * dinv[col[i]];
}

// ---------------- SpMM: y = A_hat @ z  (feature width 64) ----------------

__global__ void k_spmm_init(const float* __restrict__ dinv, const float* __restrict__ z,
                            float* __restrict__ y, int n) {
    int i = blockIdx.x * blockDim.x + threadIdx.x;
    if (i < n) {
        float d = dinv[i >> 6];               // self-loop weight = dinv^2
        y[i] = d * d * z[i];
    }
}

// One wave per edge: edge metadata forced wave-uniform (scalar loads),
// each lane moves a float2 of the 64-wide feature row (L2-resident).
__global__ __launch_bounds__(256)
void k_spmm_scatter(const int* __restrict__ row, const int* __restrict__ col,
                    const float* __restrict__ w, const float* __restrict__ z,
                    float* __restrict__ y, int e) {
    const int lane = threadIdx.x & 31;
    const int wv   = __builtin_amdgcn_readfirstlane(threadIdx.x >> 5);
    const int ei   = blockIdx.x * 8 + wv;
    if (ei < e) {
        const int   r  = row[ei];
        const int   c  = col[ei];
        const float wvv = w[ei];
        const float2 zz = *(const float2*)(z + (size_t)c * HDIM + 2 * lane);
        float* yp = y + (size_t)r * HDIM + 2 * lane;
        atomicAdd(yp,     wvv * zz.x);
        atomicAdd(yp + 1, wvv * zz.y);
    }
}

// ---------------- elementwise epilogues ----------------

__global__ void k_relu_scale(const float* __restrict__ s, const float* __restrict__ a,
                             float* __restrict__ out, int n) {
    int i = blockIdx.x * blockDim.x + threadIdx.x;
    if (i < n) {
        float v = a[0] * s[i];
        out[i] = v > 0.0f ? v : 0.0f;
    }
}

__global__ void k_het_ep(const float* __restrict__ z, const float* __restrict__ s,
                         const float* __restrict__ a, float* __restrict__ out, int n) {
    int i = blockIdx.x * blockDim.x + threadIdx.x;
    if (i < n) {
        float v = a[0] * (z[i] - s[i]);
        out[i] = v > 0.0f ? v : 0.0f;
    }
}

__global__ void k_combine(const float* __restrict__ hh, const float* __restrict__ hf,
                          const float* __restrict__ he,
                          const float* __restrict__ wh, const float* __restrict__ wf,
                          const float* __restrict__ we, float* __restrict__ out, int n) {
    int i = blockIdx.x * blockDim.x + threadIdx.x;
    if (i < n) out[i] = wh[0] * hh[i] + wf[0] * hf[i] + we[0] * he[i];
}

// ---------------- WMMA GEMM: C[M x 64] = A[M x K] @ W[K x 64] ----------------
// One wave per 16x16 tile; 4 waves/block cover a 16-row stripe of all 64 cols.
// W panel staged into LDS via async copy (GLOBAL_LOAD_ASYNC_TO_LDS_B128).
// mode: 0 = raw store, 1 = relu(alpha * c).

__global__ __launch_bounds__(128)
void k_gemm16(const float* __restrict__ A, const float* __restrict__ W,
              float* __restrict__ C, int K, int mode, const float* __restrict__ alpha_p)
{
    __shared__ __align__(16) float Wl[DIN * HDIM];   // up to 128x64 f32 = 32 KB
    const int tid = threadIdx.x;

    // Async cache->LDS staging, 16B per issue, tracked on ASYNCcnt.
    for (int i = tid * 4; i < K * HDIM; i += 128 * 4) {
        unsigned     ldsOff = (unsigned)(size_t)(&Wl[i]);   // AS3 byte offset
        const float* gsrc   = W + i;
        asm volatile("global_load_async_to_lds_b128 %0, %1, off"
                     :: "v"(ldsOff), "v"(gsrc) : "memory");
    }
    asm volatile("s_wait_asynccnt 0x0" ::: "memory");
    __syncthreads();

    const int lane = tid & 31;
    const int wv   = tid >> 5;                // n-tile index 0..3
    const int c    = lane & 15;
    const int kh   = lane >> 4;               // 0/1 half-wave selector
    const int m0   = blockIdx.x << 4;

    v8f acc = {};

    // f32 WMMA, K-step 4. A 16x4 layout: lane = M + 16*(K/2), elems = K pair.
    const float* arow = A + (size_t)(m0 + c) * K + 2 * kh;
    const float* bcol = Wl + (2 * kh) * HDIM + (wv << 4) + c;
    for (int k = 0; k < K; k += 4) {
        v2f a, b;
        a[0] = arow[k];
        a[1] = arow[k + 1];
        b[0] = bcol[k * HDIM];
        b[1] = bcol[k * HDIM + HDIM];
        acc = __builtin_amdgcn_wmma_f32_16x16x4_f32(
            false, a, false, b, (short)0, acc, false, false);
        __builtin_prefetch(arow + k + 32, 0, 3);
    }

    const float alpha = alpha_p ? alpha_p[0] : 1.0f;
    float* cp = C + (size_t)(m0 + (kh << 3)) * HDIM + (wv << 4) + c;
    #pragma unroll
    for (int r = 0; r < 8; ++r) {             // elem r -> M = m0 + r + 8*kh, N = wv*16+c
        float v = acc[r];
        if (mode) { v *= alpha; v = v > 0.0f ? v : 0.0f; }
        cp[r * HDIM] = v;
    }
}

// ---------------- launcher ----------------

extern "C" void kernel_launch(void* const* d_in, const int* in_sizes, int n_in,
                              void* d_out, int out_size, void* d_ws, size_t ws_size,
                              hipStream_t stream) {
    (void)in_sizes; (void)n_in; (void)out_size; (void)ws_size;

    const float* x       = (const float*)d_in[0];
    const int*   ei      = (const int*)d_in[1];
    const int*   row     = ei;
    const int*   col     = ei + NEDGES;
    const float* homo_W0 = (const float*)d_in[3];
    const float* homo_W1 = (const float*)d_in[4];
    const float* full_W0 = (const float*)d_in[5];
    const float* full_W1 = (const float*)d_in[6];
    const float* het_W0  = (const float*)d_in[7];
    const float* het_W1  = (const float*)d_in[8];
    const float* homo_a0 = (const float*)d_in[9];
    const float* homo_a1 = (const float*)d_in[10];
    const float* full_a0 = (const float*)d_in[11];
    const float* full_a1 = (const float*)d_in[12];
    const float* het_a0  = (const float*)d_in[13];
    const float* het_a1  = (const float*)d_in[14];
    const float* w_homo  = (const float*)d_in[15];
    const float* w_full  = (const float*)d_in[16];
    const float* w_het   = (const float*)d_in[17];

    float* out      = (float*)d_out;
    float* out_comb = out;
    float* out_homo = out + (size_t)1 * NHSZ;
    float* out_full = out + (size_t)2 * NHSZ;
    float* out_het  = out + (size_t)3 * NHSZ;

    float* ws    = (float*)d_ws;
    float* dinv  = ws;                  // N
    float* wedge = dinv + NNODES;       // E
    float* zA    = wedge + NEDGES;      // N*64  (homo z / h)
    float* zB    = zA + NHSZ;           // N*64  (het z / h)
    float* F     = zB + NHSZ;           // N*64  (full hidden)
    float* sA    = F + NHSZ;            // N*64
    float* sB    = sA + NHSZ;           // N*64
    float* T     = sB + NHSZ;           // N*64

    const int B = 256;
    const int gN  = (NNODES + B - 1) / B;
    const int gE  = (NEDGES + B - 1) / B;
    const int gNH = (NHSZ + B - 1) / B;
    const int gSc = NEDGES / 8;                 // 100000, one wave per edge
    const int gM  = NNODES / 16;                // 3125, exact

    // adjacency normalization
    k_deg_init <<<gN, B, 0, stream>>>(dinv, NNODES);
    k_deg_count<<<gE, B, 0, stream>>>(col, dinv, NEDGES);
    k_dinv     <<<gN, B, 0, stream>>>(dinv, NNODES);
    k_edge_w   <<<gE, B, 0, stream>>>(row, col, dinv, wedge, NEDGES);

    // ---- layer 0: GEMM first (commute), then 64-wide L2-resident SpMM ----
    k_gemm16<<<gM, 128, 0, stream>>>(x, homo_W0, zA, DIN, 0, nullptr);
    k_gemm16<<<gM, 128, 0, stream>>>(x, het_W0,  zB, DIN, 0, nullptr);
    k_gemm16<<<gM, 128, 0, stream>>>(x, full_W0, F,  DIN, 1, full_a0);

    k_spmm_init   <<<gNH, B, 0, stream>>>(dinv, zA, sA, NHSZ);
    k_spmm_scatter<<<gSc, B, 0, stream>>>(row, col, wedge, zA, sA, NEDGES);
    k_relu_scale  <<<gNH, B, 0, stream>>>(sA, homo_a0, zA, NHSZ);     // h_homo0 -> zA

    k_spmm_init   <<<gNH, B, 0, stream>>>(dinv, zB, sB, NHSZ);
    k_spmm_scatter<<<gSc, B, 0, stream>>>(row, col, wedge, zB, sB, NEDGES);
    k_het_ep      <<<gNH, B, 0, stream>>>(zB, sB, het_a0, zB, NHSZ);  // h_het0 -> zB

    // ---- layer 1 ----
    k_gemm16<<<gM, 128, 0, stream>>>(zA, homo_W1, sA, HDIM, 0, nullptr);
    k_spmm_init   <<<gNH, B, 0, stream>>>(dinv, sA, T, NHSZ);
    k_spmm_scatter<<<gSc, B, 0, stream>>>(row, col, wedge, sA, T, NEDGES);
    k_relu_scale  <<<gNH, B, 0, stream>>>(T, homo_a1, out_homo, NHSZ);

    k_gemm16<<<gM, 128, 0, stream>>>(zB, het_W1, sB, HDIM, 0, nullptr);
    k_spmm_init   <<<gNH, B, 0, stream>>>(dinv, sB, T, NHSZ);
    k_spmm_scatter<<<gSc, B, 0, stream>>>(row, col, wedge, sB, T, NEDGES);
    k_het_ep      <<<gNH, B, 0, stream>>>(sB, T, het_a1, out_het, NHSZ);

    k_gemm16<<<gM, 128, 0, stream>>>(F, full_W1, out_full, HDIM, 1, full_a1);

    // ---- combine ----
    k_combine<<<gNH, B, 0, stream>>>(out_homo, out_full, out_het,
                                     w_homo, w_full, w_het, out_comb, NHSZ);
}